// Conditionals_32066225832719
// MI455X (gfx1250) — compile-verified
//
#include <hip/hip_runtime.h>

typedef float v2f __attribute__((ext_vector_type(2)));
typedef float v8f __attribute__((ext_vector_type(8)));

constexpr int Nn     = 100;   // nodes
constexpr int Dn     = 32;    // DAGs
constexpr int Bn     = 8192;  // batch
constexpr int Hn     = 32;    // hidden
constexpr int K1     = 101;   // N + 1 (x input column appended)
constexpr int ROWS   = 128;   // batch rows per block (4 waves x 32 rows)
constexpr int STRIDE = 101;   // LDS row stride (odd -> conflict free)

__global__ __launch_bounds__(128) void dag_mlp_wmma(
    const float* __restrict__ x,      // (B, N)
    const float* __restrict__ A,      // (D, N, N)
    const int*   __restrict__ order,  // (D, N)
    const int*   __restrict__ do_idx_p,
    const float* __restrict__ W1,     // (N, H, N+1)
    const float* __restrict__ b1,     // (N, H)
    const float* __restrict__ W2,     // (N, H)
    const float* __restrict__ b2,     // (N,)
    float*       __restrict__ out)    // (D, B, N)
{
  __shared__ float S[ROWS * STRIDE];  // evolving outputs state, per-block

  const int d    = blockIdx.y;
  const int tile = blockIdx.x;
  const int tid  = threadIdx.x;
  const int lane = tid & 31;
  const int wave = tid >> 5;
  const int half = lane >> 4;   // 0: lanes 0-15, 1: lanes 16-31
  const int l16  = lane & 15;
  const int do_idx = do_idx_p[0];

  for (int i = tid; i < ROWS * STRIDE; i += 128) S[i] = 0.0f;
  __syncthreads();  // only barrier; afterwards each wave touches only its rows

  const int rl0 = wave * 32;          // local row base owned by this wave
  const int g0  = tile * ROWS + rl0;  // global batch row base

  const int*   ord = order + d * Nn;
  const float* Ad  = A + (size_t)d * Nn * Nn;

  for (int s = 0; s < Nn; ++s) {
    const int node = ord[s];          // wave-uniform

    if (node == do_idx) {
      // intervention: value forced to x[:, node]; mask column already zeroed
      const int   rl = rl0 + lane;
      const float xv = x[(size_t)(g0 + lane) * Nn + node];
      S[rl * STRIDE + node] = xv;
      out[((size_t)d * Bn + (size_t)(g0 + lane)) * Nn + node] = xv;
      continue;
    }

    const float* W1n  = W1 + (size_t)node * Hn * K1;
    const float  b1v0 = b1[node * Hn + l16];        // column h = l16
    const float  b1v1 = b1[node * Hn + 16 + l16];   // column h = 16 + l16

    v8f acc00, acc01, acc10, acc11;   // [m-tile][n-tile]
#pragma unroll
    for (int e = 0; e < 8; ++e) {
      acc00[e] = b1v0; acc01[e] = b1v1;
      acc10[e] = b1v0; acc11[e] = b1v1;
    }

    for (int k0 = 0; k0 < K1; k0 += 4) {   // 26 K-steps covering k = 0..103
      const int kb = k0 + half * 2;        // this half's K pair base

      // mask along K, folded into B: k<100 -> A[d][k][node]; k==100 -> 1 (x col); else 0
      const float m0 = (kb     < Nn) ? Ad[kb * Nn + node]
                                     : ((kb     == Nn) ? 1.0f : 0.0f);
      const float m1 = ((kb+1) < Nn) ? Ad[(kb + 1) * Nn + node]
                                     : (((kb+1) == Nn) ? 1.0f : 0.0f);

      // B fragments: B[k][n] = W1[node][n][k] * mask[k]; lane -> n = ntile*16 + l16
      v2f bf0, bf1;
      {
        const float w00 = (kb     < K1) ? W1n[l16 * K1 + kb]            : 0.0f;
        const float w01 = ((kb+1) < K1) ? W1n[l16 * K1 + kb + 1]        : 0.0f;
        const float w10 = (kb     < K1) ? W1n[(16 + l16) * K1 + kb]     : 0.0f;
        const float w11 = ((kb+1) < K1) ? W1n[(16 + l16) * K1 + kb + 1] : 0.0f;
        bf0[0] = w00 * m0; bf0[1] = w01 * m1;
        bf1[0] = w10 * m0; bf1[1] = w11 * m1;
      }

      // A fragments: raw ins values (mask lives in B); lane -> M = l16 of each m-tile
      v2f af0, af1;
      {
        const int r0 = rl0 + l16;        // m-tile 0 local row
        const int r1 = rl0 + 16 + l16;   // m-tile 1 local row
        af0[0] = (kb     < Nn) ? S[r0 * STRIDE + kb]
               : ((kb     == Nn) ? x[(size_t)(g0 + l16) * Nn + node] : 0.0f);
        af0[1] = ((kb+1) < Nn) ? S[r0 * STRIDE + kb + 1]
               : (((kb+1) == Nn) ? x[(size_t)(g0 + l16) * Nn + node] : 0.0f);
        af1[0] = (kb     < Nn) ? S[r1 * STRIDE + kb]
               : ((kb     == Nn) ? x[(size_t)(g0 + 16 + l16) * Nn + node] : 0.0f);
        af1[1] = ((kb+1) < Nn) ? S[r1 * STRIDE + kb + 1]
               : (((kb+1) == Nn) ? x[(size_t)(g0 + 16 + l16) * Nn + node] : 0.0f);
      }

      // fp32 WMMA: D = A(16x4) x B(4x16) + C ; exact fp32 accumulation
      acc00 = __builtin_amdgcn_wmma_f32_16x16x4_f32(false, af0, false, bf0,
                                                    (short)0, acc00, false, false);
      acc01 = __builtin_amdgcn_wmma_f32_16x16x4_f32(false, af0, false, bf1,
                                                    (short)0, acc01, false, false);
      acc10 = __builtin_amdgcn_wmma_f32_16x16x4_f32(false, af1, false, bf0,
                                                    (short)0, acc10, false, false);
      acc11 = __builtin_amdgcn_wmma_f32_16x16x4_f32(false, af1, false, bf1,
                                                    (short)0, acc11, false, false);
    }

    // Layer 2: out = leaky_relu(z) . W2[node] + b2[node]
    const float w2_0 = W2[node * Hn + l16];
    const float w2_1 = W2[node * Hn + 16 + l16];
    const float b2v  = b2[node];

#pragma unroll
    for (int mt = 0; mt < 2; ++mt) {
      v8f za = (mt == 0) ? acc00 : acc10;
      v8f zb = (mt == 0) ? acc01 : acc11;
#pragma unroll
      for (int e = 0; e < 8; ++e) {
        float h0 = za[e]; h0 = (h0 > 0.0f) ? h0 : 0.01f * h0;   // leaky_relu(0.01)
        float h1 = zb[e]; h1 = (h1 > 0.0f) ? h1 : 0.01f * h1;
        float p  = h0 * w2_0 + h1 * w2_1;                       // partial over h
        // reduce the 16 columns held across the lanes of this half
        p += __shfl_xor(p, 1, 16);
        p += __shfl_xor(p, 2, 16);
        p += __shfl_xor(p, 4, 16);
        p += __shfl_xor(p, 8, 16);
        if (l16 == e) {                                         // one writer per row
          const int   M   = half * 8 + e;                       // C-layout row
          const int   rl  = rl0 + mt * 16 + M;
          const float val = p + b2v;
          S[rl * STRIDE + node] = val;
          out[((size_t)d * Bn + (size_t)(tile * ROWS + rl)) * Nn + node] = val;
        }
      }
    }
  }
}

extern "C" void kernel_launch(void* const* d_in, const int* in_sizes, int n_in,
                              void* d_out, int out_size, void* d_ws, size_t ws_size,
                              hipStream_t stream) {
  (void)in_sizes; (void)n_in; (void)out_size; (void)d_ws; (void)ws_size;
  const float* x      = (const float*)d_in[0];
  const float* A      = (const float*)d_in[1];
  const int*   order  = (const int*)d_in[2];
  const int*   do_idx = (const int*)d_in[3];
  const float* W1     = (const float*)d_in[4];
  const float* b1     = (const float*)d_in[5];
  const float* W2     = (const float*)d_in[6];
  const float* b2     = (const float*)d_in[7];
  float*       out    = (float*)d_out;

  dim3 grid(Bn / ROWS, Dn);   // 64 batch tiles x 32 DAGs = 2048 blocks
  dag_mlp_wmma<<<grid, 128, 0, stream>>>(x, A, order, do_idx, W1, b1, W2, b2, out);
}